// ComplexCoherencyloss_83434034692663
// MI455X (gfx1250) — compile-verified
//
#include <hip/hip_runtime.h>
#include <math.h>

typedef __attribute__((ext_vector_type(2))) float v2f;
typedef __attribute__((ext_vector_type(8))) float v8f;

#define HW       512
#define OD       508          // output H'=W' = 512 - 5 + 1
#define ROWS_IN  36           // 32 output rows + 4 halo
#define XPAD     272          // 260 needed cols padded for LDS banking
#define XV       (XPAD / 4)   // 68 float4 columns
#define SQ       (ROWS_IN * XPAD)       // floats per quantity plane: 9792
#define LDS_BYTES (4 * SQ * sizeof(float))  // 156,672 B of 320 KB WGP LDS

__global__ __launch_bounds__(256) void coherency_main(
    const float* __restrict__ pr, const float* __restrict__ pi,
    const float* __restrict__ tr, const float* __restrict__ ti,
    double* __restrict__ acc)
{
    extern __shared__ float S[];      // S[q*SQ + r*XPAD + x], q = {cr, ci, pp, tt}
    __shared__ float wg_sum;

    const int tid = threadIdx.x;
    if (tid == 0) wg_sum = 0.0f;

    const int x0 = blockIdx.x * 256;  // output column base (0 or 256)
    const int y0 = blockIdx.y * 32;   // output row base
    const int b  = blockIdx.z;

    const size_t plane = (size_t)HW * HW;
    const size_t base0 = ((size_t)b * 2 + 0) * plane;   // channel 0
    const size_t base1 = ((size_t)b * 2 + 1) * plane;   // channel 1

    // ---- Phase 1: stream inputs once, fuse elementwise + channel-sum into LDS ----
    for (int idx = tid; idx < ROWS_IN * XV; idx += 256) {
        const int r  = idx / XV;
        const int xv = idx - r * XV;
        const int gy = y0 + r;
        const int gx = x0 + xv * 4;

        float4 cr = {0,0,0,0}, ci = {0,0,0,0}, pp = {0,0,0,0}, tt = {0,0,0,0};
        if (gy < HW && gx < HW) {     // gx multiple of 4, so gx<512 => gx+3<512
            const size_t o0 = base0 + (size_t)gy * HW + gx;
            const size_t o1 = base1 + (size_t)gy * HW + gx;
            const float4 pr0 = *(const float4*)(pr + o0);
            const float4 pr1 = *(const float4*)(pr + o1);
            const float4 pi0 = *(const float4*)(pi + o0);
            const float4 pi1 = *(const float4*)(pi + o1);
            const float4 tr0 = *(const float4*)(tr + o0);
            const float4 tr1 = *(const float4*)(tr + o1);
            const float4 ti0 = *(const float4*)(ti + o0);
            const float4 ti1 = *(const float4*)(ti + o1);
#define EW(c) \
            cr.c = pr0.c*tr0.c + pi0.c*ti0.c + pr1.c*tr1.c + pi1.c*ti1.c; \
            ci.c = pi0.c*tr0.c - pr0.c*ti0.c + pi1.c*tr1.c - pr1.c*ti1.c; \
            pp.c = pr0.c*pr0.c + pi0.c*pi0.c + pr1.c*pr1.c + pi1.c*pi1.c; \
            tt.c = tr0.c*tr0.c + ti0.c*ti0.c + tr1.c*tr1.c + ti1.c*ti1.c;
            EW(x) EW(y) EW(z) EW(w)
#undef EW
        }
        const int loff = r * XPAD + xv * 4;
        *(float4*)&S[0*SQ + loff] = cr;
        *(float4*)&S[1*SQ + loff] = ci;
        *(float4*)&S[2*SQ + loff] = pp;
        *(float4*)&S[3*SQ + loff] = tt;
    }
    __syncthreads();

    // ---- Phase 2: 5x5 box sum = horizontal 5-tap (VALU from LDS) then
    //      vertical 5-tap as a banded-ones matmul on the WMMA pipe:
    //      D(16x16) = sum_{t=0..4} A_t(16x4, band) x H_t(4x16)
    const int wave = tid >> 5;
    const int lane = tid & 31;            // wave32
    const int n    = lane & 15;
    const int kh   = (lane >> 4) << 1;    // K-half: 0 (lanes 0-15) or 2 (lanes 16-31)
    const int m    = n;                   // A-matrix row index for this lane

    // A band constants: A[m][k] = 1 iff m <= k <= m+4  (k = 4t + kh + j)
    v2f afrag[5];
#pragma unroll
    for (int t = 0; t < 5; ++t) {
        const int k0 = 4*t + kh;
        afrag[t].x = (k0     >= m && k0     <= m + 4) ? 1.0f : 0.0f;
        afrag[t].y = (k0 + 1 >= m && k0 + 1 <= m + 4) ? 1.0f : 0.0f;
    }

    float local = 0.0f;
    // 32 tiles per WG: 16 column-tiles x 2 row-halves; uniform per wave => EXEC all 1s
    for (int tile = wave; tile < 32; tile += 8) {
        const int half  = tile & 1;
        const int tx    = tile >> 1;
        const int xloc  = tx * 16;        // local column base (<= 240)
        const int rbase = half * 16;      // local patch-row base (0 or 16)

        v8f q[4];
#pragma unroll
        for (int qq = 0; qq < 4; ++qq) {
            const float* Sq = &S[qq * SQ];
            v8f c = {0,0,0,0,0,0,0,0};
#pragma unroll
            for (int t = 0; t < 5; ++t) {
                const int r0 = rbase + 4*t + kh;          // <= 35
                const float* p0 = &Sq[r0 * XPAD + xloc + n];
                const float* p1 = p0 + XPAD;
                v2f bfrag;                                 // B 4x16: VGPR j -> K = kh + j
                bfrag.x = p0[0] + p0[1] + p0[2] + p0[3] + p0[4];
                bfrag.y = p1[0] + p1[1] + p1[2] + p1[3] + p1[4];
                c = __builtin_amdgcn_wmma_f32_16x16x4_f32(
                        false, afrag[t], false, bfrag, (short)0, c, false, false);
            }
            q[qq] = c;
        }

        // C/D layout: VGPR v -> M = v (lanes 0-15) / M = v+8 (lanes 16-31), N = lane&15
        const int xg    = x0 + xloc + n;
        const int ybase = y0 + rbase + ((lane >> 4) << 3);
#pragma unroll
        for (int v = 0; v < 8; ++v) {
            const int yg = ybase + v;
            if (yg < OD && xg < OD) {
                const float nr = q[0][v], ni = q[1][v];
                const float dp = q[2][v], dt = q[3][v];
                local += sqrtf(nr*nr + ni*ni) / sqrtf(dp * dt);
            }
        }
    }

    atomicAdd(&wg_sum, local);            // ds_add_f32
    __syncthreads();
    if (tid == 0) atomicAdd(acc, (double)wg_sum);   // global_atomic_add_f64
}

__global__ void coherency_init(double* __restrict__ acc) { *acc = 0.0; }

__global__ void coherency_final(const double* __restrict__ acc, float* __restrict__ out)
{
    out[0] = 1.0f - (float)(*acc / (16.0 * (double)OD * (double)OD));
}

extern "C" void kernel_launch(void* const* d_in, const int* in_sizes, int n_in,
                              void* d_out, int out_size, void* d_ws, size_t ws_size,
                              hipStream_t stream)
{
    (void)in_sizes; (void)n_in; (void)out_size; (void)ws_size;
    const float* pr = (const float*)d_in[0];
    const float* pi = (const float*)d_in[1];
    const float* tr = (const float*)d_in[2];
    const float* ti = (const float*)d_in[3];
    // d_in[4] = filter_size (==5, specialized at compile time)
    double* acc = (double*)d_ws;
    float*  out = (float*)d_out;

    coherency_init<<<1, 1, 0, stream>>>(acc);
    dim3 grid(2, 16, 16);   // (column half, row strip, batch)
    coherency_main<<<grid, 256, LDS_BYTES, stream>>>(pr, pi, tr, ti, acc);
    coherency_final<<<1, 1, 0, stream>>>(acc, out);
}